// AutoencoderHybrid_65481071396421
// MI455X (gfx1250) — compile-verified
//
#include <hip/hip_runtime.h>
#include <math.h>
#include <stdint.h>

// CDNA5 / gfx1250: wave32, WMMA 16x16 shapes. V_WMMA_F32_16X16X4_F32 applies
// the fused low-nibble (qubits 0-3) gate: state viewed as (256 x 16) f32
// matrix pair (re, im) times the fused 16x16 complex unitary U16^T.
// Gate tables are staged global->LDS with GLOBAL_LOAD_ASYNC_TO_LDS_B32
// (ASYNCcnt), overlapped with the initial product-state construction.

typedef __attribute__((ext_vector_type(2))) float v2f;
typedef __attribute__((ext_vector_type(8))) float v8f;

#define NQ      12
#define NSTATE  4096      // 2^12 amplitudes
#define NROW    256       // qubits 4..11 (index bits 7..0)
#define SCOLS   257       // padded column stride (257 % 64 == 1 -> no bank conflicts)
#define NLAYERS 2
#define LATENT  4
#define HID     32
#define QBATCH  8192

// padded LDS index for flat amplitude index n = col*256 + row
#define SIDX(n) ((n) + ((n) >> 8))

// d_ws float layout
#define WS_HIGH 0                  // [2 layers][8 gates (q=4..11)][8 floats 2x2 cplx]
#define WS_LOW  128                // [2 layers][4 gates (q=0..3)][8 floats]
#define WS_BT   192                // [2 layers][3 mats: re, im, -im][256]  Bt[k*16+n]=U16[n][k]
#define WS_BT_LSTRIDE 768
#define BT_FLOATS (NLAYERS * WS_BT_LSTRIDE)   // 1536 floats = 6 KB

// ---------------------------------------------------------------------------
// Kernel 1: build Rot matrices + fused 16x16 low-nibble unitary (transposed)
// PennyLane Rot(phi,theta,omega) = RZ(omega) RY(theta) RZ(phi)
// ---------------------------------------------------------------------------
__global__ void qae_prep(const float* __restrict__ qp, float* __restrict__ ws) {
  const int t = threadIdx.x;
  if (t < NLAYERS * NQ) {
    const int l = t / NQ, i = t % NQ;
    const float phi = qp[(l * NQ + i) * 3 + 0];
    const float th  = qp[(l * NQ + i) * 3 + 1];
    const float om  = qp[(l * NQ + i) * 3 + 2];
    const float c = cosf(0.5f * th), s = sinf(0.5f * th);
    const float p = 0.5f * (phi + om), m = 0.5f * (phi - om);
    const float cp = cosf(p), sp = sinf(p), cm = cosf(m), sm = sinf(m);
    float u[8];
    u[0] =  c * cp; u[1] = -c * sp;   // U00
    u[2] = -s * cm; u[3] = -s * sm;   // U01
    u[4] =  s * cm; u[5] = -s * sm;   // U10
    u[6] =  c * cp; u[7] =  c * sp;   // U11
    float* dst = (i < 4) ? (ws + WS_LOW + (l * 4 + i) * 8)
                         : (ws + WS_HIGH + (l * 8 + (i - 4)) * 8);
    for (int j = 0; j < 8; ++j) dst[j] = u[j];
  }
  __syncthreads();
  // Fused U16[n][k] = prod_{q=0..3} R_q[bit_{3-q}(n)][bit_{3-q}(k)]; store U16^T
  for (int e = t; e < NLAYERS * 256; e += blockDim.x) {
    const int l = e >> 8;
    const int idx = e & 255;
    const int n = idx & 15;   // destination nibble (N of the WMMA B operand)
    const int k = idx >> 4;   // source nibble (K, contraction index)
    float re = 1.f, im = 0.f;
    const float* lg = ws + WS_LOW + l * 32;
    #pragma unroll
    for (int q = 0; q < 4; ++q) {
      const int a = (n >> (3 - q)) & 1;
      const int b = (k >> (3 - q)) & 1;
      const float* R = lg + q * 8 + (a * 2 + b) * 2;
      const float rr = R[0], ri = R[1];
      const float nre = re * rr - im * ri;
      const float nim = re * ri + im * rr;
      re = nre; im = nim;
    }
    float* bt = ws + WS_BT + l * WS_BT_LSTRIDE;
    bt[0 * 256 + k * 16 + n] =  re;
    bt[1 * 256 + k * 16 + n] =  im;
    bt[2 * 256 + k * 16 + n] = -im;   // pre-negated: f32 WMMA NEG only has C-negate
  }
}

// ---------------------------------------------------------------------------
// Kernel 2: one workgroup (8 wave32) per batch element; state in LDS.
// Index convention (reference flattening): qubit i <-> bit (11-i) of n.
// n = col*256 + row : col = qubits 0..3, row = qubits 4..11.
// ---------------------------------------------------------------------------
__global__ __launch_bounds__(256) void qae_sim(
    const float* __restrict__ xin, const float* __restrict__ ws,
    const float* __restrict__ W1, const float* __restrict__ b1,
    const float* __restrict__ W2, const float* __restrict__ b2,
    float* __restrict__ out) {
  __shared__ float sRe[16 * SCOLS];
  __shared__ float sIm[16 * SCOLS];
  __shared__ float sBt[BT_FLOATS];      // staged U16^T tables (both layers)
  __shared__ float sLat[LATENT];
  __shared__ float sH[HID];

  const int t = threadIdx.x;
  const int b = blockIdx.x;
  const int lane = t & 31;          // wave32
  const int wave = t >> 5;
  const int half = lane >> 4;       // WMMA lane-half
  const int ln   = lane & 15;

  __builtin_prefetch(xin + b * NQ, 0, 0);   // global_prefetch_b8

  // --- 0) async-stage the WMMA B tables into LDS (overlaps state init) ---
  // 1536 dwords / 256 threads = 6 per thread; per-lane LDS dest in VDST VGPR.
  #pragma unroll
  for (int i = 0; i < BT_FLOATS / 256; ++i) {
    const unsigned lds_off = (unsigned)(size_t)(void*)&sBt[t + i * 256];
    const uint64_t gaddr = (uint64_t)(const void*)(ws + WS_BT + t + i * 256);
    asm volatile("global_load_async_to_lds_b32 %0, %1, off"
                 :: "v"(lds_off), "v"(gaddr) : "memory");
  }

  // --- 1) initial product state: RX(x_i)|0> = [cos(x/2), -i sin(x/2)] ---
  float cc[NQ], sn[NQ];
  #pragma unroll
  for (int i = 0; i < NQ; ++i) {
    const float x = xin[b * NQ + i];
    cc[i] = cosf(0.5f * x);
    sn[i] = sinf(0.5f * x);
  }
  // common product over qubits 0..7 (they live in the bits of t: n = t*16+k)
  float magH = 1.f;
  int popH = 0;
  #pragma unroll
  for (int i = 0; i < 8; ++i) {
    const int bit = (t >> (7 - i)) & 1;
    magH *= bit ? sn[i] : cc[i];
    popH += bit;
  }
  #pragma unroll
  for (int k = 0; k < 16; ++k) {
    float m = magH;
    int pop = popH;
    #pragma unroll
    for (int i = 8; i < NQ; ++i) {
      const int bit = (k >> (11 - i)) & 1;
      m *= bit ? sn[i] : cc[i];
      pop += bit;
    }
    // amp = (-i)^pop * m
    const int n = t * 16 + k;
    float re, im;
    if (pop & 1) { re = 0.f; im = (pop & 2) ? m : -m; }
    else         { im = 0.f; re = (pop & 2) ? -m : m; }
    sRe[SIDX(n)] = re;
    sIm[SIDX(n)] = im;
  }

  // --- CNOT chain source indices (state-independent; same for both layers) ---
  int srcIdx[16];
  #pragma unroll
  for (int k = 0; k < 16; ++k) {
    int src = t * 16 + k;
    #pragma unroll
    for (int c = 10; c >= 0; --c)          // invert: each CNOT is involutive
      src ^= ((src >> (11 - c)) & 1) << (10 - c);
    srcIdx[k] = SIDX(src);
  }

  asm volatile("s_wait_asynccnt 0" ::: "memory");   // B tables resident in LDS
  __syncthreads();

  for (int l = 0; l < NLAYERS; ++l) {
    // --- 2a) fused Rot gates on qubits 0..3 via f32 WMMA:  S' = S * U16^T ---
    const float* bt = sBt + l * WS_BT_LSTRIDE;
    v2f Bre[4], Bim[4], Bimn[4];    // B (4x16): lane L, vgpr v -> K=4j+2*half+v, N=ln
    #pragma unroll
    for (int j = 0; j < 4; ++j) {
      const int k0 = 4 * j + 2 * half;
      Bre[j].x  = bt[0 * 256 + (k0 + 0) * 16 + ln];
      Bre[j].y  = bt[0 * 256 + (k0 + 1) * 16 + ln];
      Bim[j].x  = bt[1 * 256 + (k0 + 0) * 16 + ln];
      Bim[j].y  = bt[1 * 256 + (k0 + 1) * 16 + ln];
      Bimn[j].x = bt[2 * 256 + (k0 + 0) * 16 + ln];
      Bimn[j].y = bt[2 * 256 + (k0 + 1) * 16 + ln];
    }
    #pragma unroll
    for (int tile = 0; tile < 2; ++tile) {
      const int base = wave * 32 + tile * 16;    // 16 state-rows per tile
      v8f dr = {0, 0, 0, 0, 0, 0, 0, 0};
      v8f di = {0, 0, 0, 0, 0, 0, 0, 0};
      #pragma unroll
      for (int j = 0; j < 4; ++j) {              // K = 16 in 4 chunks of 4
        const int k0 = 4 * j + 2 * half;
        const int row = base + ln;               // A: lane L -> M=ln, K=k0+v
        v2f are, aim;
        are.x = sRe[(k0 + 0) * SCOLS + row];
        are.y = sRe[(k0 + 1) * SCOLS + row];
        aim.x = sIm[(k0 + 0) * SCOLS + row];
        aim.y = sIm[(k0 + 1) * SCOLS + row];
        // Dr = Sr*Ur + Si*(-Ui) ; Di = Sr*Ui + Si*Ur
        dr = __builtin_amdgcn_wmma_f32_16x16x4_f32(false, are, false, Bre[j],  (short)0, dr, false, false);
        dr = __builtin_amdgcn_wmma_f32_16x16x4_f32(false, aim, false, Bimn[j], (short)0, dr, false, false);
        di = __builtin_amdgcn_wmma_f32_16x16x4_f32(false, are, false, Bim[j],  (short)0, di, false, false);
        di = __builtin_amdgcn_wmma_f32_16x16x4_f32(false, aim, false, Bre[j],  (short)0, di, false, false);
      }
      #pragma unroll
      for (int v = 0; v < 8; ++v) {              // D: vgpr v -> M=v+8*half, N=ln
        const int row = base + v + 8 * half;
        sRe[ln * SCOLS + row] = dr[v];           // padded stride -> conflict-free
        sIm[ln * SCOLS + row] = di[v];
      }
    }
    __syncthreads();

    // --- 2b) Rot gates on qubits 4..11: pairwise complex 2x2 updates ---
    for (int g = 0; g < 8; ++g) {
      const float* U = ws + WS_HIGH + (l * 8 + g) * 8;  // uniform -> s_load
      const float u00r = U[0], u00i = U[1], u01r = U[2], u01i = U[3];
      const float u10r = U[4], u10i = U[5], u11r = U[6], u11i = U[7];
      const int beta = 7 - g;                   // qubit q=4+g -> bit 11-q of n
      #pragma unroll
      for (int pp = 0; pp < 8; ++pp) {
        const int p = t * 8 + pp;               // pair id, 0..2047
        const int n0 = ((p >> beta) << (beta + 1)) | (p & ((1 << beta) - 1));
        const int i0 = SIDX(n0), i1 = SIDX(n0 | (1 << beta));
        const float a0r = sRe[i0], a0i = sIm[i0];
        const float a1r = sRe[i1], a1i = sIm[i1];
        sRe[i0] = u00r * a0r - u00i * a0i + u01r * a1r - u01i * a1i;
        sIm[i0] = u00r * a0i + u00i * a0r + u01r * a1i + u01i * a1r;
        sRe[i1] = u10r * a0r - u10i * a0i + u11r * a1r - u11i * a1i;
        sIm[i1] = u10r * a0i + u10i * a0r + u11r * a1i + u11i * a1r;
      }
      __syncthreads();
    }

    // --- 2c) CNOT chain 0->1 ... 10->11 composed into one gather pass ---
    float tre[16], tim[16];
    #pragma unroll
    for (int k = 0; k < 16; ++k) {
      tre[k] = sRe[srcIdx[k]];
      tim[k] = sIm[srcIdx[k]];
    }
    __syncthreads();
    #pragma unroll
    for (int k = 0; k < 16; ++k) {
      const int y = SIDX(t * 16 + k);
      sRe[y] = tre[k];
      sIm[y] = tim[k];
    }
    __syncthreads();
  }

  // --- 3) latents: <Z_i> = sum_n |amp(n)|^2 * (1 - 2*bit_{11-i}(n)), i<4 ---
  if (t < LATENT) sLat[t] = 0.f;
  __syncthreads();
  float part[LATENT] = {0.f, 0.f, 0.f, 0.f};
  #pragma unroll
  for (int k = 0; k < 16; ++k) {
    const int n = t * 16 + k;
    const int s = SIDX(n);
    const float pr = sRe[s] * sRe[s] + sIm[s] * sIm[s];
    #pragma unroll
    for (int i = 0; i < LATENT; ++i)
      part[i] += ((n >> (11 - i)) & 1) ? -pr : pr;
  }
  #pragma unroll
  for (int i = 0; i < LATENT; ++i) atomicAdd(&sLat[i], part[i]);  // ds_add_f32
  __syncthreads();

  // --- 4) MLP: relu(L @ W1^T + b1) @ W2^T + b2 (tiny; scalar epilogue) ---
  if (t < HID) {
    float acc = b1[t];
    #pragma unroll
    for (int k = 0; k < LATENT; ++k) acc += sLat[k] * W1[t * LATENT + k];
    sH[t] = acc > 0.f ? acc : 0.f;
  }
  __syncthreads();
  if (t < NQ) {
    float acc = b2[t];
    #pragma unroll
    for (int j = 0; j < HID; ++j) acc += sH[j] * W2[t * HID + j];
    out[b * NQ + t] = acc;
  }
}

extern "C" void kernel_launch(void* const* d_in, const int* in_sizes, int n_in,
                              void* d_out, int out_size, void* d_ws, size_t ws_size,
                              hipStream_t stream) {
  const float* xin = (const float*)d_in[0];   // (8192, 12)
  const float* qp  = (const float*)d_in[1];   // (2, 12, 3)
  const float* W1  = (const float*)d_in[2];   // (32, 4)
  const float* b1  = (const float*)d_in[3];   // (32,)
  const float* W2  = (const float*)d_in[4];   // (12, 32)
  const float* b2  = (const float*)d_in[5];   // (12,)
  float* out = (float*)d_out;                 // (8192, 12)
  float* ws  = (float*)d_ws;                  // ~7 KB used

  qae_prep<<<1, 256, 0, stream>>>(qp, ws);
  qae_sim<<<QBATCH, 256, 0, stream>>>(xin, ws, W1, b1, W2, b2, out);
}